// MPN_6545530159157
// MI455X (gfx1250) — compile-verified
//
#include <hip/hip_runtime.h>
#include <math.h>

#define Bsz 4
#define Nn  10000
#define Ee  160000
#define Ff  128
#define FEe 16
#define NT  8          // 128 / 16 output column tiles
#define KS_MSG 68      // K = 272 = 68 * 4
#define KS_UPD 64      // K = 256 = 64 * 4

typedef __attribute__((ext_vector_type(2))) float v2f;
typedef __attribute__((ext_vector_type(8))) float v8f;

__device__ __forceinline__ v8f wmma4(v2f a, v2f b, v8f c) {
  // D = A(16x4 f32) * B(4x16 f32) + C(16x16 f32), full f32 precision
  return __builtin_amdgcn_wmma_f32_16x16x4_f32(false, a, false, b, (short)0, c,
                                               false, false);
}

__device__ __forceinline__ float gelu_exact(float x) {
  return 0.5f * x * (1.0f + erff(x * 0.7071067811865476f));
}

__device__ __forceinline__ void atomic_add_f32(float* p, float v) {
  (void)__hip_atomic_fetch_add(p, v, __ATOMIC_RELAXED, __HIP_MEMORY_SCOPE_AGENT);
}

// One k-step: A fragment from global (per-lane direct gather), 8 B fragments
// from pre-swizzled LDS, 8 WMMAs into the 16x128 accumulator.
__device__ __forceinline__ void kstep(const float* __restrict__ ap,
                                      const float* __restrict__ wp,
                                      v8f acc[NT]) {
  v2f a = *(const v2f*)ap;                           // global_load_b64
#pragma unroll
  for (int nt = 0; nt < NT; ++nt) {
    v2f bf = *(const v2f*)(wp + nt * 64);            // ds_load(_2addr)_b64
    acc[nt] = wmma4(a, bf, acc[nt]);
  }
}

// ---------------------------------------------------------------- zero ws
__global__ void zero_kernel(float4* __restrict__ p, int n4) {
  int i = blockIdx.x * blockDim.x + threadIdx.x;
  int st = gridDim.x * blockDim.x;
  float4 z = make_float4(0.f, 0.f, 0.f, 0.f);
  for (; i < n4; i += st) p[i] = z;
}

// ------------------------------------------------------- edge message pass
// One wave = one tile of 16 edges. W_msg pre-swizzled into LDS as WMMA B
// fragments: for k-step kk, col-tile nt, lane l, j in {0,1}:
//   ldsW[kk*512 + nt*64 + l*2 + j] = W[4*kk + 2*(l>=16) + j][nt*16 + (l&15)]
__global__ __launch_bounds__(256) void msg_kernel(
    const float* __restrict__ nodes, const float* __restrict__ ef,
    const int*   __restrict__ edges, const float* __restrict__ ew,
    const float* __restrict__ edrop, const float* __restrict__ W,
    const float* __restrict__ bias,  const float* __restrict__ g_,
    const float* __restrict__ be_,   float* __restrict__ agg) {
  __shared__ float ldsW[KS_MSG * NT * 64];   // 139264 B
  __shared__ float ldsP[3 * Ff];             // bias | gamma | beta

  const int tid = threadIdx.x;
  for (int i = tid; i < Ff; i += blockDim.x) {
    ldsP[i] = bias[i]; ldsP[Ff + i] = g_[i]; ldsP[2 * Ff + i] = be_[i];
  }
  for (int i = tid; i < KS_MSG * NT * 64; i += blockDim.x) {
    int j    = i & 1;
    int lane = (i >> 1) & 31;
    int nt   = (i >> 6) & 7;
    int kk   = i >> 9;
    int k    = 4 * kk + ((lane >> 4) << 1) + j;
    int n    = nt * 16 + (lane & 15);
    ldsW[i]  = W[k * Ff + n];
  }
  __syncthreads();

  const int wave = tid >> 5;
  const int lane = tid & 31;
  const int hf   = lane >> 4;     // which K-half of the A fragment this lane holds
  const int r    = lane & 15;     // edge row within tile
  const int wavesPerBlk = blockDim.x >> 5;
  const int nTiles = Bsz * (Ee / 16);

  for (int t = blockIdx.x * wavesPerBlk + wave; t < nTiles;
       t += gridDim.x * wavesPerBlk) {
    const int b  = t / (Ee / 16);
    const int et = t - b * (Ee / 16);
    const int e  = et * 16 + r;
    const size_t eb = (size_t)b * Ee + e;
    const int src = edges[eb * 2 + 0];
    const int dst = edges[eb * 2 + 1];
    const float wgt = ew[eb] * edrop[eb * 2 + 1];
    const float* srcRow = nodes + ((size_t)b * Nn + src) * Ff + 2 * hf;
    const float* dstRow = nodes + ((size_t)b * Nn + dst) * Ff + 2 * hf;
    const float* efRow  = ef + eb * FEe + 2 * hf;

    v8f acc[NT] = {};
    const float* wp = ldsW + lane * 2;
    // K regions are uniform per k-step across both lane halves, so the loop
    // splits into three branch-free phases (src | dst | edge-features).
#pragma unroll 4
    for (int kk = 0; kk < 32; ++kk) { kstep(srcRow + 4 * kk, wp, acc); wp += 512; }
#pragma unroll 4
    for (int kk = 0; kk < 32; ++kk) { kstep(dstRow + 4 * kk, wp, acc); wp += 512; }
#pragma unroll
    for (int kk = 0; kk < 4;  ++kk) { kstep(efRow  + 4 * kk, wp, acc); wp += 512; }

    // bias + exact GELU
#pragma unroll
    for (int nt = 0; nt < NT; ++nt) {
      const float bb = ldsP[nt * 16 + r];
#pragma unroll
      for (int v = 0; v < 8; ++v) acc[nt][v] = gelu_exact(acc[nt][v] + bb);
    }

    // LayerNorm stats: row (v + 8*hf) lives in this lane's 16-lane half
    float mu[8], rs[8];
#pragma unroll
    for (int v = 0; v < 8; ++v) {
      float s = 0.f, q = 0.f;
#pragma unroll
      for (int nt = 0; nt < NT; ++nt) { float x = acc[nt][v]; s += x; q += x * x; }
#pragma unroll
      for (int m = 1; m <= 8; m <<= 1) {
        s += __shfl_xor(s, m, 32);
        q += __shfl_xor(q, m, 32);
      }
      const float mean = s * (1.0f / Ff);
      mu[v] = mean;
      rs[v] = rsqrtf(q * (1.0f / Ff) - mean * mean + 1e-3f);
    }

    // normalize, scale, scatter-add to destination nodes
#pragma unroll
    for (int v = 0; v < 8; ++v) {
      const int rr = v + 8 * hf;               // edge row of this value
      const float wrow = __shfl(wgt, rr, 32);  // lanes 0..15 hold rows 0..15
      const int   drow = __shfl(dst, rr, 32);
      float* aggRow = agg + ((size_t)b * Nn + drow) * Ff;
#pragma unroll
      for (int nt = 0; nt < NT; ++nt) {
        const int col = nt * 16 + r;
        const float val =
            ((acc[nt][v] - mu[v]) * rs[v] * ldsP[Ff + col] + ldsP[2 * Ff + col]) * wrow;
        atomic_add_f32(aggRow + col, val);
      }
    }
  }
}

// ---------------------------------------------------------- node update
__global__ __launch_bounds__(256) void upd_kernel(
    const float* __restrict__ nodes, const float* __restrict__ agg,
    const float* __restrict__ W,     const float* __restrict__ bias,
    const float* __restrict__ g_,    const float* __restrict__ be_,
    float* __restrict__ out) {
  __shared__ float ldsW[KS_UPD * NT * 64];   // 131072 B
  __shared__ float ldsP[3 * Ff];

  const int tid = threadIdx.x;
  for (int i = tid; i < Ff; i += blockDim.x) {
    ldsP[i] = bias[i]; ldsP[Ff + i] = g_[i]; ldsP[2 * Ff + i] = be_[i];
  }
  for (int i = tid; i < KS_UPD * NT * 64; i += blockDim.x) {
    int j    = i & 1;
    int lane = (i >> 1) & 31;
    int nt   = (i >> 6) & 7;
    int kk   = i >> 9;
    int k    = 4 * kk + ((lane >> 4) << 1) + j;
    int n    = nt * 16 + (lane & 15);
    ldsW[i]  = W[k * Ff + n];
  }
  __syncthreads();

  const int wave = tid >> 5;
  const int lane = tid & 31;
  const int hf   = lane >> 4;
  const int r    = lane & 15;
  const int wavesPerBlk = blockDim.x >> 5;
  const int nTiles = Bsz * (Nn / 16);

  for (int t = blockIdx.x * wavesPerBlk + wave; t < nTiles;
       t += gridDim.x * wavesPerBlk) {
    const int b  = t / (Nn / 16);
    const int nb = (t - b * (Nn / 16)) * 16;   // node base of tile
    const float* nRow = nodes + ((size_t)b * Nn + nb + r) * Ff + 2 * hf;
    const float* aRow = agg   + ((size_t)b * Nn + nb + r) * Ff + 2 * hf;

    v8f acc[NT] = {};
    const float* wp = ldsW + lane * 2;
#pragma unroll 4
    for (int kk = 0; kk < 32; ++kk) { kstep(nRow + 4 * kk, wp, acc); wp += 512; }
#pragma unroll 4
    for (int kk = 0; kk < 32; ++kk) { kstep(aRow + 4 * kk, wp, acc); wp += 512; }

#pragma unroll
    for (int nt = 0; nt < NT; ++nt) {
      const float bb = ldsP[nt * 16 + r];
#pragma unroll
      for (int v = 0; v < 8; ++v) acc[nt][v] = gelu_exact(acc[nt][v] + bb);
    }

    float mu[8], rs[8];
#pragma unroll
    for (int v = 0; v < 8; ++v) {
      float s = 0.f, q = 0.f;
#pragma unroll
      for (int nt = 0; nt < NT; ++nt) { float x = acc[nt][v]; s += x; q += x * x; }
#pragma unroll
      for (int m = 1; m <= 8; m <<= 1) {
        s += __shfl_xor(s, m, 32);
        q += __shfl_xor(q, m, 32);
      }
      const float mean = s * (1.0f / Ff);
      mu[v] = mean;
      rs[v] = rsqrtf(q * (1.0f / Ff) - mean * mean + 1e-3f);
    }

#pragma unroll
    for (int v = 0; v < 8; ++v) {
      const int rr = v + 8 * hf;
      float* oRow = out + ((size_t)b * Nn + nb + rr) * Ff;
#pragma unroll
      for (int nt = 0; nt < NT; ++nt) {
        const int col = nt * 16 + r;
        oRow[col] = (acc[nt][v] - mu[v]) * rs[v] * ldsP[Ff + col] + ldsP[2 * Ff + col];
      }
    }
  }
}

extern "C" void kernel_launch(void* const* d_in, const int* in_sizes, int n_in,
                              void* d_out, int out_size, void* d_ws, size_t ws_size,
                              hipStream_t stream) {
  (void)in_sizes; (void)n_in; (void)out_size; (void)ws_size;
  const float* nodes = (const float*)d_in[0];
  const float* ef    = (const float*)d_in[1];
  const int*   edges = (const int*)d_in[2];
  const float* ew    = (const float*)d_in[3];
  const float* edrop = (const float*)d_in[4];
  const float* Wm    = (const float*)d_in[5];
  const float* bm    = (const float*)d_in[6];
  const float* gm    = (const float*)d_in[7];
  const float* bem   = (const float*)d_in[8];
  const float* Wu    = (const float*)d_in[9];
  const float* bu    = (const float*)d_in[10];
  const float* gu    = (const float*)d_in[11];
  const float* beu   = (const float*)d_in[12];
  float* out = (float*)d_out;
  float* agg = (float*)d_ws;   // [B, N, F] f32 aggregate

  const int n4 = (Bsz * Nn * Ff) / 4;
  zero_kernel<<<2560, 256, 0, stream>>>((float4*)agg, n4);
  msg_kernel<<<1024, 256, 0, stream>>>(nodes, ef, edges, ew, edrop,
                                       Wm, bm, gm, bem, agg);
  upd_kernel<<<160, 256, 0, stream>>>(nodes, agg, Wu, bu, gu, beu, out);
}